// GCNConv_53334903882610
// MI455X (gfx1250) — compile-verified
//
#include <hip/hip_runtime.h>
#include <hip/hip_bf16.h>

typedef __attribute__((ext_vector_type(2))) float v2f;
typedef __attribute__((ext_vector_type(8))) float v8f;

#define N_NODES 100000
#define N_EDGES 1600000
#define D_FEAT  64
#define UNITS   64

// ---------------------------------------------------------------------------
// Phase 1: in-degree via float atomics (deg[target] += 1)
// ---------------------------------------------------------------------------
__global__ __launch_bounds__(256) void gcn_degree_kernel(
    const int* __restrict__ target, float* __restrict__ deg, int nEdges) {
  int e = blockIdx.x * blockDim.x + threadIdx.x;
  if (e < nEdges) atomicAdd(&deg[target[e]], 1.0f);
}

// ---------------------------------------------------------------------------
// Phase 2: dinv = rsqrt(deg)  (deg==0 -> inf, matching jax.lax.rsqrt)
// ---------------------------------------------------------------------------
__global__ __launch_bounds__(256) void gcn_rsqrt_kernel(
    const float* __restrict__ deg, float* __restrict__ dinv, int n) {
  int i = blockIdx.x * blockDim.x + threadIdx.x;
  if (i < n) dinv[i] = rsqrtf(deg[i]);
}

// ---------------------------------------------------------------------------
// Phase 3: pooled[tgt] += dinv[src] * x[src]   (one wave32 per edge,
// 2 floats per lane => 64 features; coalesced float2 reads, f32 atomics to L2)
// ---------------------------------------------------------------------------
__global__ __launch_bounds__(256) void gcn_scatter_kernel(
    const float* __restrict__ x, const int* __restrict__ src,
    const int* __restrict__ tgt, const float* __restrict__ dinv,
    float* __restrict__ pooled, int nEdges) {
  int wave = (blockIdx.x * blockDim.x + threadIdx.x) >> 5;
  int lane = threadIdx.x & 31;
  if (wave >= nEdges) return;
  int s = src[wave];
  int t = tgt[wave];
  float d = dinv[s];
  const float2* xs = (const float2*)(x + (size_t)s * D_FEAT);
  float2 v = xs[lane];
  float* p = pooled + (size_t)t * D_FEAT + 2 * lane;
  atomicAdd(p + 0, v.x * d);
  atomicAdd(p + 1, v.y * d);
}

// ---------------------------------------------------------------------------
// Phase 4: out = relu( (dinv[:,None] * pooled) @ W + b )
// fp32 WMMA (V_WMMA_F32_16X16X4_F32), 4 waves/block, 16 rows x 64 cols/block.
// W (16KB) + scaled A tile (4KB) staged in LDS.
// ---------------------------------------------------------------------------
__global__ __launch_bounds__(128) void gcn_gemm_wmma_kernel(
    const float* __restrict__ pooled, const float* __restrict__ W,
    const float* __restrict__ bias, const float* __restrict__ dinv,
    float* __restrict__ out) {
  __shared__ float sW[D_FEAT * UNITS];   // 64x64 row-major (k, n)
  __shared__ float sA[16 * D_FEAT];      // 16x64 row-major (m, k), pre-scaled

  const int tid = threadIdx.x;
  const int rowBase = blockIdx.x * 16;

  // Cooperative load of W: 1024 float4, 8 per thread (coalesced B128 loads).
  {
    const float4* Wv = (const float4*)W;
    float4* sWv = (float4*)sW;
    for (int i = tid; i < (D_FEAT * UNITS) / 4; i += 128) sWv[i] = Wv[i];
  }
  // Cooperative load of A tile, fused with dinv row scaling.
  {
    const float4* Av = (const float4*)(pooled + (size_t)rowBase * D_FEAT);
    float4* sAv = (float4*)sA;
    for (int i = tid; i < (16 * D_FEAT) / 4; i += 128) {
      int row = i / (D_FEAT / 4);
      float d = dinv[rowBase + row];
      float4 v = Av[i];
      v.x *= d; v.y *= d; v.z *= d; v.w *= d;
      sAv[i] = v;
    }
  }
  __syncthreads();

  const int lane  = tid & 31;
  const int wave  = tid >> 5;
  const int nBase = wave * 16;     // each wave owns a 16-wide N tile
  const int h     = lane >> 4;     // lane half: selects K sub-pair / M half
  const int mn    = lane & 15;     // A row (m) / B,D column (n)

  v8f acc = {};
  #pragma unroll
  for (int kk = 0; kk < D_FEAT; kk += 4) {
    const int k0 = kk + 2 * h;
    v2f a, b;
    // A 16x4 f32 fragment: lanes 0-15 hold K={kk,kk+1}, lanes 16-31 K={kk+2,kk+3}
    a.x = sA[mn * D_FEAT + k0];
    a.y = sA[mn * D_FEAT + k0 + 1];
    // B 4x16 f32 fragment: row K striped across lanes within a VGPR
    b.x = sW[(k0)     * UNITS + nBase + mn];
    b.y = sW[(k0 + 1) * UNITS + nBase + mn];
    // D = A*B + C : v_wmma_f32_16x16x4_f32
    acc = __builtin_amdgcn_wmma_f32_16x16x4_f32(
        /*neg_a=*/false, a, /*neg_b=*/false, b,
        /*c_mod=*/(short)0, acc, /*reuse_a=*/false, /*reuse_b=*/false);
  }

  // Epilogue: bias + ReLU, D layout VGPR j -> row (j + 8h), col mn.
  const float bn = bias[nBase + mn];
  #pragma unroll
  for (int j = 0; j < 8; ++j) {
    const int m = j + 8 * h;
    float r = acc[j] + bn;
    out[(size_t)(rowBase + m) * UNITS + nBase + mn] = fmaxf(r, 0.0f);
  }
}

// ---------------------------------------------------------------------------
// Launcher
// ---------------------------------------------------------------------------
extern "C" void kernel_launch(void* const* d_in, const int* in_sizes, int n_in,
                              void* d_out, int out_size, void* d_ws, size_t ws_size,
                              hipStream_t stream) {
  const float* x      = (const float*)d_in[0];
  const float* W      = (const float*)d_in[1];
  const float* b      = (const float*)d_in[2];
  const int*   source = (const int*)d_in[3];
  const int*   target = (const int*)d_in[4];
  float*       out    = (float*)d_out;

  float* pooled = (float*)d_ws;                            // N*64 floats
  float* deg    = pooled + (size_t)N_NODES * D_FEAT;       // N floats
  float* dinv   = deg + N_NODES;                           // N floats

  // Zero accumulators every call (graph-capture safe, deterministic).
  hipMemsetAsync(pooled, 0, (size_t)N_NODES * D_FEAT * sizeof(float), stream);
  hipMemsetAsync(deg,    0, (size_t)N_NODES * sizeof(float), stream);

  gcn_degree_kernel<<<(N_EDGES + 255) / 256, 256, 0, stream>>>(target, deg, N_EDGES);
  gcn_rsqrt_kernel<<<(N_NODES + 255) / 256, 256, 0, stream>>>(deg, dinv, N_NODES);
  // one wave32 per edge, 8 waves per 256-thread block
  gcn_scatter_kernel<<<(N_EDGES + 7) / 8, 256, 0, stream>>>(x, source, target, dinv,
                                                            pooled, N_EDGES);
  // 16 rows per block, 100000/16 = 6250 blocks exactly
  gcn_gemm_wmma_kernel<<<N_NODES / 16, 128, 0, stream>>>(pooled, W, b, dinv, out);
}